// ScaleInvariantLogLoss_29085518529245
// MI455X (gfx1250) — compile-verified
//
#include <hip/hip_runtime.h>

typedef float v2f __attribute__((ext_vector_type(2)));
typedef float v4f __attribute__((ext_vector_type(4)));
typedef float v8f __attribute__((ext_vector_type(8)));

#define EPS  1e-6f
#define LN2  0.69314718055994530942f

// ---------------------------------------------------------------------------
// Kernel 1: streaming partials. Each block covers 4096 elements of one sample.
// Accumulation of ld and ld^2 runs on the WMMA pipe (f32 16x16x4, B = ones):
// D[m,n] += sum_k A[m,k]  -> sum of all 64 wave-wide A values is recoverable
// with one shfl_xor(16); lane placement of values is irrelevant for a sum.
// ---------------------------------------------------------------------------
__global__ __launch_bounds__(256) void silog_partials(
    const float* __restrict__ pred, const float* __restrict__ targ,
    float* __restrict__ ws, int blocksPerSample, int nBlocks)
{
    const int b   = blockIdx.x;
    const int s   = b / blocksPerSample;          // sample
    const int c   = b % blocksPerSample;          // chunk within sample
    const int tid = threadIdx.x;

    const v4f* p4 = (const v4f*)pred;
    const v4f* t4 = (const v4f*)targ;
    // 307200 elems/sample = 76800 float4; 4096 elems/block = 1024 float4
    const long base = (long)s * 76800 + (long)c * 1024;

    v8f acc_s = {};            // accumulates sum(ld2)
    v8f acc_q = {};            // accumulates sum(ld2^2)
    v2f ones;  ones[0] = 1.0f; ones[1] = 1.0f;
    float cnt = 0.0f;

#pragma unroll
    for (int it = 0; it < 4; ++it) {
        const long i = base + (long)it * 256 + tid;
        v4f p = __builtin_nontemporal_load(&p4[i]);   // global_load_b128 NT
        v4f t = __builtin_nontemporal_load(&t4[i]);

        float ld[4];
#pragma unroll
        for (int k = 0; k < 4; ++k) {
            const bool m = t[k] > 0.0f;
            // log2 native; ln2 scaling folded into the final reduction
            const float d = __log2f(p[k] + EPS) - __log2f(t[k] + EPS);
            ld[k] = m ? d : 0.0f;                 // v_cndmask kills any NaN
            cnt  += m ? 1.0f : 0.0f;
        }

        v2f a0, a1, q0, q1;
        a0[0] = ld[0];        a0[1] = ld[1];
        a1[0] = ld[2];        a1[1] = ld[3];
        q0[0] = ld[0]*ld[0];  q0[1] = ld[1]*ld[1];
        q1[0] = ld[2]*ld[2];  q1[1] = ld[3]*ld[3];

        acc_s = __builtin_amdgcn_wmma_f32_16x16x4_f32(false, a0, false, ones,
                                                      (short)0, acc_s, false, false);
        acc_s = __builtin_amdgcn_wmma_f32_16x16x4_f32(false, a1, false, ones,
                                                      (short)0, acc_s, false, false);
        acc_q = __builtin_amdgcn_wmma_f32_16x16x4_f32(false, q0, false, ones,
                                                      (short)0, acc_q, false, false);
        acc_q = __builtin_amdgcn_wmma_f32_16x16x4_f32(false, q1, false, ones,
                                                      (short)0, acc_q, false, false);
    }

    // Extract wave totals from the 16x16 accumulators.
    // Lanes 0-15 hold rows M=0..7 in VGPRs 0..7 (any column n is identical
    // since B is all ones); lanes 16-31 hold rows M=8..15.
    float ssum = acc_s[0]+acc_s[1]+acc_s[2]+acc_s[3]
               + acc_s[4]+acc_s[5]+acc_s[6]+acc_s[7];
    float qsum = acc_q[0]+acc_q[1]+acc_q[2]+acc_q[3]
               + acc_q[4]+acc_q[5]+acc_q[6]+acc_q[7];
    ssum += __shfl_xor(ssum, 16, 32);
    qsum += __shfl_xor(qsum, 16, 32);
#pragma unroll
    for (int off = 16; off > 0; off >>= 1)
        cnt += __shfl_xor(cnt, off, 32);

    __shared__ float sh[3][8];
    const int w    = tid >> 5;
    const int lane = tid & 31;
    if (lane == 0) { sh[0][w] = ssum; sh[1][w] = qsum; sh[2][w] = cnt; }
    __syncthreads();
    if (tid == 0) {
        float S = 0.f, Q = 0.f, C = 0.f;
#pragma unroll
        for (int i = 0; i < 8; ++i) { S += sh[0][i]; Q += sh[1][i]; C += sh[2][i]; }
        ws[b]              = S;
        ws[nBlocks + b]    = Q;
        ws[2*nBlocks + b]  = C;
    }
}

// ---------------------------------------------------------------------------
// Kernel 2: one 1024-thread block, one wave32 per sample. Deterministic.
// ---------------------------------------------------------------------------
__global__ __launch_bounds__(1024) void silog_final(
    const float* __restrict__ ws, float* __restrict__ out,
    int blocksPerSample, int nBlocks, int B)
{
    const int tid  = threadIdx.x;
    const int w    = tid >> 5;    // sample index
    const int lane = tid & 31;

    __shared__ float per[32];

    float S = 0.f, Q = 0.f, C = 0.f;
    if (w < B) {
        for (int j = lane; j < blocksPerSample; j += 32) {
            const int idx = w * blocksPerSample + j;
            S += ws[idx];
            Q += ws[nBlocks + idx];
            C += ws[2*nBlocks + idx];
        }
    }
#pragma unroll
    for (int off = 16; off > 0; off >>= 1) {
        S += __shfl_xor(S, off, 32);
        Q += __shfl_xor(Q, off, 32);
        C += __shfl_xor(C, off, 32);
    }

    if (w < B && lane == 0) {
        const float sum_ld = S * LN2;              // log2 -> ln
        const float sum_sq = Q * (LN2 * LN2);
        const float safe   = fmaxf(C, 1.0f);
        const float mld    = sum_ld / safe;
        const float msq    = sum_sq / safe;
        per[w] = (C > 0.0f) ? (msq - 0.5f * mld * mld) : 0.0f;
    }
    __syncthreads();

    if (tid == 0) {
        float tot = 0.f;
        for (int i = 0; i < B; ++i) tot += per[i];
        out[0] = tot / (float)B;
    }
}

extern "C" void kernel_launch(void* const* d_in, const int* in_sizes, int n_in,
                              void* d_out, int out_size, void* d_ws, size_t ws_size,
                              hipStream_t stream)
{
    (void)in_sizes; (void)n_in; (void)out_size; (void)ws_size;

    const float* pred = (const float*)d_in[0];
    const float* targ = (const float*)d_in[1];
    float*       out  = (float*)d_out;
    float*       ws   = (float*)d_ws;     // needs 3 * 2400 * 4 B = 28.8 KB

    const int B   = 32;
    const int BPS = 75;                   // 307200 / 4096 elements per block
    const int NB  = B * BPS;              // 2400 blocks, exact coverage

    silog_partials<<<NB, 256, 0, stream>>>(pred, targ, ws, BPS, NB);
    silog_final<<<1, 1024, 0, stream>>>(ws, out, BPS, NB, B);
}